// Attention_987842478321
// MI455X (gfx1250) — compile-verified
//
#include <hip/hip_runtime.h>
#include <math.h>

// ---------------------------------------------------------------------------
// Attention: out=(B,1,D) tanh((softmax(q·ctxT)·ctx ++ q) @ W_outT + b), attn
// B=64, N=2048, D=1024, f32.
// Roofline: context = 512MB >> L2(192MB). Single HBM pass over context via
// online softmax (~22us @ 23.3TB/s). Final GEMM (64x1024x2048, W L2-resident)
// uses V_WMMA_F32_16X16X4_F32.
// ---------------------------------------------------------------------------

typedef __attribute__((ext_vector_type(2))) float v2f;
typedef __attribute__((ext_vector_type(8))) float v8f;

#define BB 64
#define NN 2048
#define DD 1024
#define EE 2048      // 2*D
#define SPLIT 8      // row-chunks per batch in phase 1
#define ROWS_PER_CHUNK (NN / SPLIT)   // 256
#define ROWS_PER_WAVE (ROWS_PER_CHUNK / 8) // 32

// ---------------------------------------------------------------------------
// Phase 1: one block per (batch, chunk). 8 waves, each wave owns 32 context
// rows. Per row: coalesced float4 loads, wave-reduced dot with q, online
// softmax update of (m, l, acc[1024-distributed]). Cross-wave merge via LDS
// f32 atomics. Emits: raw scores -> attn region (normalized later), partial
// mix/m/l -> workspace.
// ---------------------------------------------------------------------------
__global__ __launch_bounds__(256) void attn_phase1(
    const float* __restrict__ qin,    // (B,1,D)   d_in[0]
    const float* __restrict__ ctx,    // (B,N,D)   d_in[1]
    float* __restrict__ attn,         // (B,N) raw scores for now
    float* __restrict__ pmix,         // (B,SPLIT,D) unnormalized partial mix
    float* __restrict__ pm,           // (B,SPLIT) chunk max
    float* __restrict__ pl)           // (B,SPLIT) chunk sum (rel. chunk max)
{
  __shared__ float smix[DD];
  __shared__ float wm[8];
  __shared__ float wl[8];

  const int b     = blockIdx.x;
  const int chunk = blockIdx.y;
  const int tid   = threadIdx.x;
  const int wave  = tid >> 5;
  const int lane  = tid & 31;

  for (int d = tid; d < DD; d += 256) smix[d] = 0.0f;

  // q registers: q[jj] covers d = 128*jj + 4*lane .. +3 (coalesced layout)
  const float* qp = qin + (size_t)b * DD;
  float4 q[8];
#pragma unroll
  for (int jj = 0; jj < 8; ++jj)
    q[jj] = *reinterpret_cast<const float4*>(qp + 128 * jj + 4 * lane);

  float m = -__builtin_huge_valf();
  float l = 0.0f;
  float4 acc[8];
#pragma unroll
  for (int jj = 0; jj < 8; ++jj) acc[jj] = make_float4(0.f, 0.f, 0.f, 0.f);

  const int n0 = chunk * ROWS_PER_CHUNK + wave * ROWS_PER_WAVE;
  const float* rp = ctx + ((size_t)b * NN + n0) * DD;

  for (int i = 0; i < ROWS_PER_WAVE; ++i) {
    if (i + 1 < ROWS_PER_WAVE)
      __builtin_prefetch(rp + DD + 4 * lane, 0, 0);   // global_prefetch_b8

    float4 v[8];
#pragma unroll
    for (int jj = 0; jj < 8; ++jj)
      v[jj] = *reinterpret_cast<const float4*>(rp + 128 * jj + 4 * lane);

    float s = 0.f;
#pragma unroll
    for (int jj = 0; jj < 8; ++jj)
      s += q[jj].x * v[jj].x + q[jj].y * v[jj].y +
           q[jj].z * v[jj].z + q[jj].w * v[jj].w;
#pragma unroll
    for (int off = 16; off > 0; off >>= 1) s += __shfl_xor(s, off, 32);

    // online softmax (branch is wave-uniform: s identical on all lanes)
    float w;
    if (s > m) {
      const float scale = __expf(m - s);   // m==-inf first time -> 0
      l *= scale;
#pragma unroll
      for (int jj = 0; jj < 8; ++jj) {
        acc[jj].x *= scale; acc[jj].y *= scale;
        acc[jj].z *= scale; acc[jj].w *= scale;
      }
      m = s;
      w = 1.0f;
    } else {
      w = __expf(s - m);
    }
    l += w;
#pragma unroll
    for (int jj = 0; jj < 8; ++jj) {
      acc[jj].x += w * v[jj].x; acc[jj].y += w * v[jj].y;
      acc[jj].z += w * v[jj].z; acc[jj].w += w * v[jj].w;
    }
    if (lane == 0) attn[(size_t)b * NN + n0 + i] = s;  // raw score, fixed later
    rp += DD;
  }

  if (lane == 0) { wm[wave] = m; wl[wave] = l; }
  __syncthreads();

  float Mc = -__builtin_huge_valf();
#pragma unroll
  for (int i = 0; i < 8; ++i) Mc = fmaxf(Mc, wm[i]);
  float Lc = 0.f;
#pragma unroll
  for (int i = 0; i < 8; ++i) Lc += wl[i] * __expf(wm[i] - Mc);
  const float factor = __expf(m - Mc);   // rescale this wave's acc to Mc

#pragma unroll
  for (int jj = 0; jj < 8; ++jj) {
    const int d = 128 * jj + 4 * lane;
    atomicAdd(&smix[d + 0], acc[jj].x * factor);
    atomicAdd(&smix[d + 1], acc[jj].y * factor);
    atomicAdd(&smix[d + 2], acc[jj].z * factor);
    atomicAdd(&smix[d + 3], acc[jj].w * factor);
  }
  __syncthreads();

  float* pout = pmix + ((size_t)b * SPLIT + chunk) * DD;
  for (int d = tid; d < DD; d += 256) pout[d] = smix[d];
  if (tid == 0) {
    pm[b * SPLIT + chunk] = Mc;
    pl[b * SPLIT + chunk] = Lc;
  }
}

// ---------------------------------------------------------------------------
// Phase 1b: per batch, merge SPLIT chunk partials -> global (M, L); build
// combined = [mix, q]; normalize attn in place (raw scores already there).
// ---------------------------------------------------------------------------
__global__ __launch_bounds__(256) void attn_combine(
    const float* __restrict__ qin,
    const float* __restrict__ pmix,
    const float* __restrict__ pm,
    const float* __restrict__ pl,
    float* __restrict__ attn,         // raw scores in, softmax out
    float* __restrict__ combined)     // (B, 2D)
{
  const int b   = blockIdx.x;
  const int tid = threadIdx.x;

  float M = -__builtin_huge_valf();
#pragma unroll
  for (int s = 0; s < SPLIT; ++s) M = fmaxf(M, pm[b * SPLIT + s]);
  float fac[SPLIT];
  float L = 0.f;
#pragma unroll
  for (int s = 0; s < SPLIT; ++s) {
    fac[s] = __expf(pm[b * SPLIT + s] - M);
    L += pl[b * SPLIT + s] * fac[s];
  }
  const float invL = 1.0f / L;

  const float* pbase = pmix + (size_t)b * SPLIT * DD;
  for (int d = tid; d < DD; d += 256) {
    float a = 0.f;
#pragma unroll
    for (int s = 0; s < SPLIT; ++s) a += pbase[s * DD + d] * fac[s];
    combined[(size_t)b * EE + d]      = a * invL;
    combined[(size_t)b * EE + DD + d] = qin[(size_t)b * DD + d];
  }

  for (int n = tid; n < NN; n += 256) {
    const float sraw = attn[(size_t)b * NN + n];
    attn[(size_t)b * NN + n] = __expf(sraw - M) * invL;
  }
}

// ---------------------------------------------------------------------------
// Phase 2: out[b][d] = tanh( sum_e combined[b][e]*W[d][e] + bias[d] )
// GEMM M=64 N=1024 K=2048 via V_WMMA_F32_16X16X4_F32.
// 256 tiles of 16x16; block = 8 waves = one tile, K split 8 ways per wave
// (64 chained WMMAs each), LDS reduction, fused tanh+bias epilogue.
// Fragment layout per ISA 7.12.2:
//   A 16x4 : lane<16 -> {A[l][k0],A[l][k0+1]}, lane>=16 -> {A[l-16][k0+2],..+3}
//   B 4x16 : same pattern sourced from W rows (B[k][n] = W[n][k])
//   C 16x16: VGPR r -> M = r + 8*(lane>=16), N = lane&15
// ---------------------------------------------------------------------------
__global__ __launch_bounds__(256) void attn_phase2(
    const float* __restrict__ combined, // (B, 2D)
    const float* __restrict__ W,        // (D, 2D)
    const float* __restrict__ bias,     // (D)
    float* __restrict__ out)            // (B, D)
{
  __shared__ float cred[8][256];

  const int tid   = threadIdx.x;
  const int wave  = tid >> 5;
  const int lane  = tid & 31;
  const int tileM = blockIdx.x >> 6;   // 0..3   (rows of combined, step 16)
  const int tileN = blockIdx.x & 63;   // 0..63  (cols of out, step 16)

  const int r16  = lane & 15;
  const int koff = (lane >> 4) * 2;            // 0 or 2
  const float* aptr = combined + (size_t)(tileM * 16 + r16) * EE + koff;
  const float* bptr = W        + (size_t)(tileN * 16 + r16) * EE + koff;

  const int kbase = wave * (EE / 8);           // 256 K per wave

  v8f c = {0.f, 0.f, 0.f, 0.f, 0.f, 0.f, 0.f, 0.f};
#pragma unroll 4
  for (int k0 = kbase; k0 < kbase + EE / 8; k0 += 4) {
    v2f a  = *reinterpret_cast<const v2f*>(aptr + k0);
    v2f bb = *reinterpret_cast<const v2f*>(bptr + k0);
    c = __builtin_amdgcn_wmma_f32_16x16x4_f32(
        /*neg_a=*/false, a, /*neg_b=*/false, bb,
        /*c_mod=*/(short)0, c, /*reuse_a=*/false, /*reuse_b=*/false);
  }

#pragma unroll
  for (int r = 0; r < 8; ++r) cred[wave][lane * 8 + r] = c[r];
  __syncthreads();

  // one thread per accumulator slot: sum 8 K-partials, epilogue
  {
    const int idx    = tid;          // = lane_o*8 + r
    const int lane_o = idx >> 3;
    const int r      = idx & 7;
    float acc = 0.f;
#pragma unroll
    for (int w8 = 0; w8 < 8; ++w8) acc += cred[w8][idx];

    const int rowm = r + ((lane_o >> 4) << 3);          // 0..15
    const int row  = tileM * 16 + rowm;                 // batch index
    const int dcol = tileN * 16 + (lane_o & 15);        // output feature
    out[(size_t)row * DD + dcol] = tanhf(acc + bias[dcol]);
  }
}

// ---------------------------------------------------------------------------
extern "C" void kernel_launch(void* const* d_in, const int* in_sizes, int n_in,
                              void* d_out, int out_size, void* d_ws, size_t ws_size,
                              hipStream_t stream) {
  const float* qin  = (const float*)d_in[0];  // output (B,1,D)
  const float* ctx  = (const float*)d_in[1];  // context (B,N,D)
  const float* W    = (const float*)d_in[2];  // W_out (D,2D)
  const float* bias = (const float*)d_in[3];  // b_out (D)

  float* out  = (float*)d_out;                // (B,D)    = 65536 floats
  float* attn = out + BB * DD;                // (B,N)    = 131072 floats

  // workspace layout
  float* combined = (float*)d_ws;                         // B*2D   = 128K floats
  float* pmix     = combined + (size_t)BB * EE;           // B*SPLIT*D = 512K floats
  float* pm       = pmix + (size_t)BB * SPLIT * DD;       // B*SPLIT
  float* pl       = pm + BB * SPLIT;                      // B*SPLIT

  dim3 g1(BB, SPLIT);                                     // 512 workgroups
  attn_phase1<<<g1, 256, 0, stream>>>(qin, ctx, attn, pmix, pm, pl);
  attn_combine<<<BB, 256, 0, stream>>>(qin, pmix, pm, pl, attn, combined);
  attn_phase2<<<256, 256, 0, stream>>>(combined, W, bias, out);
}